// ListFoldExponentialLoss_35098472742912
// MI455X (gfx1250) — compile-verified
//
#include <hip/hip_runtime.h>
#include <cstdint>
#include <cstddef>

// Problem constants (match reference)
#define BROWS   512
#define NCOLS   16384
#define PHALF   (NCOLS / 2)
#define NTHREADS 1024
#define EPSF    1e-9f

// ---- CDNA5 async global->LDS staging (guarded so compile never breaks) ----
#if defined(__has_builtin)
#  if __has_builtin(__builtin_amdgcn_global_load_async_to_lds_b128)
#    define HAVE_ASYNC_B128 1
#  endif
#  if __has_builtin(__builtin_amdgcn_global_load_async_to_lds_b32)
#    define HAVE_ASYNC_B32 1
#  endif
#  if __has_builtin(__builtin_amdgcn_s_wait_asynccnt)
#    define HAVE_WAIT_ASYNC 1
#  endif
#endif

// Builtin pointee types: b128 wants int4*, b32 wants int* (as1 global / as3 LDS).
typedef int v4i __attribute__((ext_vector_type(4)));

// CK-style address-space launder: round-trip through uintptr_t; LLVM's
// InferAddressSpaces folds the inttoptr back into an addrspacecast.
#define GLB_V4(p) ((__attribute__((address_space(1))) v4i*)(uintptr_t)(p))
#define LDS_V4(p) ((__attribute__((address_space(3))) v4i*)(uintptr_t)(p))
#define GLB_I(p)  ((__attribute__((address_space(1))) int*)(uintptr_t)(p))
#define LDS_I(p)  ((__attribute__((address_space(3))) int*)(uintptr_t)(p))

// ---------------------------------------------------------------------------
// Per-row kernel: sort row by y_true desc (NaNs last), scans, pair loss.
// Dynamic LDS layout (bytes):
//   [0      , 64K)  vbuf : f32[16384]  y_pred values, carried through sort
//   [64K    ,128K)  kbuf : u32[16384]  sort keys;   reused as ebuf (E scan)
//   [128K   ,192K)  nbuf : f32[16384]  y_true raw;  reused as En scan
//   [192K   , ...)  partial[1024], wsum[32], icnt[1]
// ---------------------------------------------------------------------------
__global__ __launch_bounds__(NTHREADS) void row_loss_kernel(
    const float* __restrict__ y_pred, const float* __restrict__ y_true,
    float* __restrict__ row_loss, int* __restrict__ row_cnt)
{
    extern __shared__ unsigned char smem_raw[];
    float*    vbuf    = (float*)(smem_raw);
    unsigned* kbuf    = (unsigned*)(smem_raw + 64 * 1024);
    float*    ebuf    = (float*)(smem_raw + 64 * 1024);   // alias of kbuf
    float*    nbuf    = (float*)(smem_raw + 128 * 1024);
    float*    partial = (float*)(smem_raw + 192 * 1024);  // [NTHREADS]
    float*    wsum    = partial + NTHREADS;               // [32]
    int*      icnt    = (int*)(wsum + 32);

    const int tid = threadIdx.x;
    const int row = blockIdx.x;
    const float* yp = y_pred + (size_t)row * NCOLS;
    const float* yt = y_true + (size_t)row * NCOLS;

    // ---- Stage both rows into LDS via CDNA5 async-to-LDS (ASYNCcnt) ----
#if defined(HAVE_ASYNC_B128)
    for (int i = tid * 4; i < NCOLS; i += NTHREADS * 4) {
        __builtin_amdgcn_global_load_async_to_lds_b128(GLB_V4(yp + i), LDS_V4(&vbuf[i]), 0, 0);
        __builtin_amdgcn_global_load_async_to_lds_b128(GLB_V4(yt + i), LDS_V4(&nbuf[i]), 0, 0);
    }
#  if defined(HAVE_WAIT_ASYNC)
    __builtin_amdgcn_s_wait_asynccnt(0);
#  endif
#elif defined(HAVE_ASYNC_B32)
    for (int i = tid; i < NCOLS; i += NTHREADS) {
        __builtin_amdgcn_global_load_async_to_lds_b32(GLB_I(yp + i), LDS_I(&vbuf[i]), 0, 0);
        __builtin_amdgcn_global_load_async_to_lds_b32(GLB_I(yt + i), LDS_I(&nbuf[i]), 0, 0);
    }
#  if defined(HAVE_WAIT_ASYNC)
    __builtin_amdgcn_s_wait_asynccnt(0);
#  endif
#else
    for (int i = tid; i < NCOLS; i += NTHREADS) {
        vbuf[i] = yp[i];
        nbuf[i] = yt[i];
    }
#endif
    __syncthreads();

    // ---- Build order-preserving uint keys; key==0 <=> invalid (NaN) ----
    // Valid floats map to k in (0, 0xFFFFFFFF]; descending key sort ==
    // descending y_true with NaNs last (matches -inf sort key in reference).
    for (int i = tid; i < NCOLS; i += NTHREADS) {
        unsigned u  = __float_as_uint(nbuf[i]);
        bool     nn = (u & 0x7FFFFFFFu) > 0x7F800000u;  // isnan
        unsigned k  = (u & 0x80000000u) ? ~u : (u | 0x80000000u);
        kbuf[i] = nn ? 0u : k;
    }
    __syncthreads();

    // ---- Bitonic sort (descending by key), values move with keys ----
    for (unsigned kk = 2; kk <= NCOLS; kk <<= 1) {
        for (unsigned j = kk >> 1; j > 0; j >>= 1) {
            for (unsigned i = tid; i < NCOLS; i += NTHREADS) {
                unsigned ixj = i ^ j;
                if (ixj > i) {
                    unsigned ka = kbuf[i], kb = kbuf[ixj];
                    bool up = ((i & kk) == 0);
                    bool sw = up ? (ka < kb) : (ka > kb);  // descending net
                    if (sw) {
                        kbuf[i] = kb; kbuf[ixj] = ka;
                        float va = vbuf[i]; vbuf[i] = vbuf[ixj]; vbuf[ixj] = va;
                    }
                }
            }
            __syncthreads();
        }
    }

    // ---- n_valid = count of nonzero keys ----
    if (tid == 0) *icnt = 0;
    __syncthreads();
    {
        int c = 0;
        for (int i = tid; i < NCOLS; i += NTHREADS) c += (kbuf[i] != 0u);
        atomicAdd(icnt, c);
    }
    __syncthreads();
    const int nv = *icnt;

    // ---- exp(s), exp(-s), masked; overwrite keys with e (validity consumed) ----
    for (int i = tid; i < NCOLS; i += NTHREADS) {
        bool  valid = (kbuf[i] != 0u);
        float s     = vbuf[i];
        ebuf[i] = valid ? expf(s)  : 0.0f;
        nbuf[i] = valid ? expf(-s) : 0.0f;
    }
    __syncthreads();

    // ---- Inclusive block scans of ebuf and nbuf (16 elems/thread) ----
    const int CH = NCOLS / NTHREADS;  // 16
    for (int pass = 0; pass < 2; ++pass) {
        float* arr = pass ? nbuf : ebuf;
        int base = tid * CH;
        float s = 0.f;
        for (int j2 = 0; j2 < CH; ++j2) s += arr[base + j2];
        partial[tid] = s;
        __syncthreads();
        if (tid < 32) {  // per-wave totals
            float t = 0.f;
            for (int j2 = 0; j2 < 32; ++j2) t += partial[tid * 32 + j2];
            wsum[tid] = t;
        }
        __syncthreads();
        if (tid == 0) {  // exclusive scan of 32 wave totals
            float t = 0.f;
            for (int j2 = 0; j2 < 32; ++j2) { float v = wsum[j2]; wsum[j2] = t; t += v; }
        }
        __syncthreads();
        if (tid < 32) {  // exclusive per-thread offsets
            float t = wsum[tid];
            for (int j2 = 0; j2 < 32; ++j2) {
                float v = partial[tid * 32 + j2]; partial[tid * 32 + j2] = t; t += v;
            }
        }
        __syncthreads();
        {
            float t = partial[tid];
            for (int j2 = 0; j2 < CH; ++j2) { t += arr[base + j2]; arr[base + j2] = t; }
        }
        __syncthreads();
    }

    // ---- Pair losses: i in [0, nv/2), mirror jc = nv-1-i ----
    float acc = 0.f;
    const int nvHalf = nv >> 1;
    for (int i = tid; i < PHALF; i += NTHREADS) {
        if (i < nvHalf) {
            int   jc     = nv - 1 - i;          // in [1, N-1], no clip needed
            float s_top  = vbuf[i];
            float s_bot  = vbuf[jc];
            float Ci     = (i > 0) ? ebuf[i - 1] : 0.f;
            float Cni    = (i > 0) ? nbuf[i - 1] : 0.f;
            float sum_e  = ebuf[jc] - Ci;
            float sum_en = nbuf[jc] - Cni;
            float L      = (float)(nv - 2 * i);
            float arg    = sum_e * sum_en - L + EPSF;
            acc += -(s_top - s_bot - logf(arg));
        }
    }

    // ---- Block reduce -> per-row loss + counted flag ----
    partial[tid] = acc;
    __syncthreads();
    for (int sft = NTHREADS / 2; sft > 0; sft >>= 1) {
        if (tid < sft) partial[tid] += partial[tid + sft];
        __syncthreads();
    }
    if (tid == 0) {
        row_loss[row] = partial[0];
        row_cnt[row]  = (nvHalf >= 1) ? 1 : 0;
    }
}

// ---------------------------------------------------------------------------
// Deterministic final reduction over rows (no float atomics).
// ---------------------------------------------------------------------------
__global__ __launch_bounds__(BROWS) void finalize_kernel(
    const float* __restrict__ row_loss, const int* __restrict__ row_cnt,
    float* __restrict__ out)
{
    __shared__ float sl[BROWS];
    __shared__ int   sc[BROWS];
    int t = threadIdx.x;
    sl[t] = row_cnt[t] ? row_loss[t] : 0.f;
    sc[t] = row_cnt[t];
    __syncthreads();
    for (int s = BROWS / 2; s > 0; s >>= 1) {
        if (t < s) { sl[t] += sl[t + s]; sc[t] += sc[t + s]; }
        __syncthreads();
    }
    if (t == 0) out[0] = (sc[0] > 0) ? (sl[0] / (float)sc[0]) : 0.f;
}

extern "C" void kernel_launch(void* const* d_in, const int* in_sizes, int n_in,
                              void* d_out, int out_size, void* d_ws, size_t ws_size,
                              hipStream_t stream)
{
    (void)in_sizes; (void)n_in; (void)out_size; (void)ws_size;
    const float* y_pred = (const float*)d_in[0];
    const float* y_true = (const float*)d_in[1];

    float* row_loss = (float*)d_ws;
    int*   row_cnt  = (int*)((char*)d_ws + BROWS * sizeof(float));

    // 192 KB row buffers + scan scratch — fits CDNA5's 320 KB/WGP LDS.
    size_t shmem = 192 * 1024 + (NTHREADS + 32) * sizeof(float) + 16;

    row_loss_kernel<<<BROWS, NTHREADS, shmem, stream>>>(y_pred, y_true, row_loss, row_cnt);
    finalize_kernel<<<1, BROWS, 0, stream>>>(row_loss, row_cnt, (float*)d_out);
}